// Encoder_3745211482435
// MI455X (gfx1250) — compile-verified
//
#include <hip/hip_runtime.h>
#include <hip/hip_bf16.h>

typedef float v2f __attribute__((ext_vector_type(2)));
typedef float v8f __attribute__((ext_vector_type(8)));

#define F 128  // feature width (in == hid == 128 in the reference)

// ---------------------------------------------------------------------------
// Degree / normalization kernels
// ---------------------------------------------------------------------------
__global__ __launch_bounds__(256) void k_deg_init(float* __restrict__ deg, int N) {
    int i = blockIdx.x * blockDim.x + threadIdx.x;
    if (i < N) deg[i] = 1.0f;  // self-loop contributes 1 to every node's degree
}

__global__ __launch_bounds__(256) void k_deg_count(const int* __restrict__ dst,
                                                   float* __restrict__ deg, int E) {
    int e = blockIdx.x * blockDim.x + threadIdx.x;
    if (e < E) atomicAdd(&deg[dst[e]], 1.0f);
}

__global__ __launch_bounds__(256) void k_deg_rsqrt(float* __restrict__ deg, int N) {
    int i = blockIdx.x * blockDim.x + threadIdx.x;
    if (i < N) deg[i] = rsqrtf(deg[i]);  // deg >= 1 always (self-loops)
}

// ---------------------------------------------------------------------------
// GEMM: H[N x 128] = X[N x 128] @ W[128 x 128], fp32 via V_WMMA_F32_16X16X4_F32
// Block = 256 threads = 8 waves. Block b covers rows [16b, 16b+16); wave w
// covers cols [16w, 16w+16). K loop = 128 in steps of 4 -> 32 WMMAs per wave.
// Fragment layouts per CDNA5 ISA 7.12.2:
//   A 16x4 (fp32): lane holds M = lane%16, K = 2*(lane/16)+{0,1} in 2 VGPRs
//   B 4x16 (fp32): lane holds N = lane%16, K = 2*(lane/16)+{0,1} in 2 VGPRs
//   C/D 16x16:     VGPR r holds M = r + 8*(lane/16), N = lane%16
// ---------------------------------------------------------------------------
__global__ __launch_bounds__(256) void k_gemm128(const float* __restrict__ X,
                                                 const float* __restrict__ W,
                                                 float* __restrict__ H) {
    const int lane = threadIdx.x & 31;
    const int wave = threadIdx.x >> 5;          // 0..7 -> column tile
    const int row0 = blockIdx.x << 4;           // 16 rows per block
    const int n    = (wave << 4) + (lane & 15); // output column of this lane
    const int m    = row0 + (lane & 15);        // A-fragment row of this lane
    const int koff = (lane >> 4) << 1;          // 0 or 2

    const float* xrow = X + (size_t)m * F;
    v8f acc = {};
#pragma unroll
    for (int k = 0; k < F; k += 4) {
        v2f a = *(const v2f*)(xrow + k + koff);            // A[m][k+koff], A[m][k+koff+1]
        v2f b;
        b.x = W[(size_t)(k + koff + 0) * F + n];           // B[k+koff+0][n]
        b.y = W[(size_t)(k + koff + 1) * F + n];           // B[k+koff+1][n]
        acc = __builtin_amdgcn_wmma_f32_16x16x4_f32(
            /*neg_a=*/false, a, /*neg_b=*/false, b,
            /*c_mod=*/(short)0, acc, /*reuse_a=*/false, /*reuse_b=*/false);
    }
    const int mbase = row0 + ((lane >> 4) << 3);           // +8 for upper lane half
#pragma unroll
    for (int r = 0; r < 8; ++r)
        H[(size_t)(mbase + r) * F + n] = acc[r];
}

// ---------------------------------------------------------------------------
// Zero an N*128 float buffer (float4 stores)
// ---------------------------------------------------------------------------
__global__ __launch_bounds__(256) void k_zero(float4* __restrict__ p, size_t nvec) {
    size_t i = (size_t)blockIdx.x * blockDim.x + threadIdx.x;
    if (i < nvec) p[i] = make_float4(0.f, 0.f, 0.f, 0.f);
}

// ---------------------------------------------------------------------------
// Scatter-add: one wave per (edge or self-loop). Lane l moves 4 consecutive
// floats of the 128-wide row: float4 gather from H[src], 4 f32 atomics into
// OUT[dst]. All buffers are L2-resident (~77 MB total << 192 MB L2).
// Index e in [0,E) -> edge list; e in [E, E+N) -> self loop on node e-E.
// ---------------------------------------------------------------------------
__global__ __launch_bounds__(256) void k_scatter(const float* __restrict__ H,
                                                 const int* __restrict__ src,
                                                 const int* __restrict__ dst,
                                                 const float* __restrict__ dinv,
                                                 float* __restrict__ OUT,
                                                 int E, int N) {
    const int lane = threadIdx.x & 31;
    const long long e = (long long)blockIdx.x * 8 + (threadIdx.x >> 5);
    if (e >= (long long)E + N) return;
    int s, d;
    if (e < E) { s = src[e]; d = dst[e]; }
    else       { s = d = (int)(e - E); }
    const float norm = dinv[s] * dinv[d];
    const float4 hv = *(const float4*)(H + (size_t)s * F + (lane << 2));
    float* o = OUT + (size_t)d * F + (lane << 2);
    atomicAdd(o + 0, hv.x * norm);
    atomicAdd(o + 1, hv.y * norm);
    atomicAdd(o + 2, hv.z * norm);
    atomicAdd(o + 3, hv.w * norm);
}

// ---------------------------------------------------------------------------
// out[i] = relu(in[i] + bias[i % 128])  (in == out allowed)
// ---------------------------------------------------------------------------
__global__ __launch_bounds__(256) void k_bias_relu(const float* __restrict__ in,
                                                   const float* __restrict__ bias,
                                                   float* __restrict__ out,
                                                   size_t total) {
    size_t i = (size_t)blockIdx.x * blockDim.x + threadIdx.x;
    if (i < total) {
        float v = in[i] + bias[i & (F - 1)];
        out[i] = v > 0.f ? v : 0.f;
    }
}

// ---------------------------------------------------------------------------
// Host-side orchestration (graph-capture safe: launches only, all on stream)
// ---------------------------------------------------------------------------
extern "C" void kernel_launch(void* const* d_in, const int* in_sizes, int n_in,
                              void* d_out, int out_size, void* d_ws, size_t ws_size,
                              hipStream_t stream) {
    const float* x   = (const float*)d_in[0];
    const float* W1  = (const float*)d_in[1];
    const float* b1  = (const float*)d_in[2];
    const float* W2  = (const float*)d_in[3];
    const float* b2  = (const float*)d_in[4];
    const int*   ei  = (const int*)d_in[5];   // edge_index [2, E] row-major

    const int N = in_sizes[0] / F;            // 50000 (multiple of 16)
    const int E = in_sizes[5] / 2;            // 800000
    const int* srcp = ei;
    const int* dstp = ei + E;

    // Workspace layout: dinv[N] (256B-aligned pad) | bufA[N*128] | bufB[N*128]
    float* dinv = (float*)d_ws;
    size_t dinv_bytes = (((size_t)N * sizeof(float)) + 255) & ~(size_t)255;
    float* bufA = (float*)((char*)d_ws + dinv_bytes);
    float* bufB = bufA + (size_t)N * F;
    float* out  = (float*)d_out;

    const size_t total = (size_t)N * F;
    const size_t nvec  = total / 4;
    const int nodeBlocks = (N + 255) / 256;
    const int edgeBlocks = (E + 255) / 256;
    const int elemBlocks = (int)((total + 255) / 256);
    const int vecBlocks  = (int)((nvec + 255) / 256);
    const int mtiles     = N / 16;                          // 3125
    const int scatBlocks = (int)(((long long)E + N + 7) / 8);

    // --- symmetric normalization factors: dinv = rsqrt(1 + indegree) ---
    k_deg_init <<<nodeBlocks, 256, 0, stream>>>(dinv, N);
    k_deg_count<<<edgeBlocks, 256, 0, stream>>>(dstp, dinv, E);
    k_deg_rsqrt<<<nodeBlocks, 256, 0, stream>>>(dinv, N);

    // --- layer 1: h1 = relu( A_hat @ (x @ W1) + b1 ) ---
    k_gemm128  <<<mtiles, 256, 0, stream>>>(x, W1, bufA);
    k_zero     <<<vecBlocks, 256, 0, stream>>>((float4*)bufB, nvec);
    k_scatter  <<<scatBlocks, 256, 0, stream>>>(bufA, srcp, dstp, dinv, bufB, E, N);
    k_bias_relu<<<elemBlocks, 256, 0, stream>>>(bufB, b1, bufA, total);

    // --- layer 2: out = relu( A_hat @ (h1 @ W2) + b2 ) ---
    k_gemm128  <<<mtiles, 256, 0, stream>>>(bufA, W2, bufB);
    k_zero     <<<vecBlocks, 256, 0, stream>>>((float4*)out, nvec);
    k_scatter  <<<scatBlocks, 256, 0, stream>>>(bufB, srcp, dstp, dinv, out, E, N);
    k_bias_relu<<<elemBlocks, 256, 0, stream>>>(out, b2, out, total);
}